// Attention_7593502179762
// MI455X (gfx1250) — compile-verified
//
#include <hip/hip_runtime.h>
#include <hip/hip_bf16.h>
#include <math.h>

#define B_   64
#define S_   2048
#define H_   1024
#define C_   8              // S-chunks per batch
#define R_   (S_ / C_)      // 256 rows per chunk
#define TPB  256

typedef __attribute__((ext_vector_type(2))) float v2f;
typedef __attribute__((ext_vector_type(4))) float v4f;
typedef __attribute__((ext_vector_type(8))) float v8f;

// workspace layout (in floats)
#define WS_M    0
#define WS_L    (B_ * C_)
#define WS_ACC  (2 * B_ * C_)
#define WS_COMB (WS_ACC + (size_t)B_ * C_ * H_)   // combined [B, 2H]

// ---------------------------------------------------------------------------
// Kernel 1: fused scores + online softmax + weighted accumulation.
// One block per (batch, chunk). Thread t owns columns 4t..4t+3.
// lstm_output is streamed exactly once with non-temporal loads.
// ---------------------------------------------------------------------------
__global__ __launch_bounds__(TPB) void attn_pass1(
    const float* __restrict__ lstm, const float* __restrict__ hidden,
    float* __restrict__ ws)
{
  __shared__ float part[8][TPB];   // per-row dot partials
  __shared__ float sc[8];          // 8 reduced scores per group

  const int t    = threadIdx.x;
  const int lane = t & 31;
  const int wid  = t >> 5;
  const int b    = blockIdx.x / C_;
  const int c    = blockIdx.x % C_;
  const int s0   = c * R_;

  const v4f hreg = *(const v4f*)(hidden + (size_t)b * H_ + 4 * t);
  const float* base = lstm + ((size_t)b * S_ + s0) * H_ + 4 * t;

  v4f  acc   = {0.f, 0.f, 0.f, 0.f};
  float m_run = -INFINITY;
  float l_run = 0.f;

  for (int g = 0; g < R_ / 8; ++g) {
    v4f row[8];
#pragma unroll
    for (int r = 0; r < 8; ++r) {
      const v4f* p = (const v4f*)(base + (size_t)(g * 8 + r) * H_);
      row[r] = __builtin_nontemporal_load(p);   // stream: don't pollute L2
    }
#pragma unroll
    for (int r = 0; r < 8; ++r) {
      part[r][t] = row[r].x * hreg.x + row[r].y * hreg.y +
                   row[r].z * hreg.z + row[r].w * hreg.w;
    }
    __syncthreads();
    // wave `wid` reduces row `wid`: 256 partials -> 1 score
    float v = 0.f;
#pragma unroll
    for (int k = 0; k < 8; ++k) v += part[wid][lane + 32 * k];
#pragma unroll
    for (int off = 16; off > 0; off >>= 1) v += __shfl_xor(v, off, 32);
    if (lane == 0) sc[wid] = v;
    __syncthreads();
    // online softmax update (identical across all threads in the block)
#pragma unroll
    for (int r = 0; r < 8; ++r) {
      const float s     = sc[r];
      const float m_new = fmaxf(m_run, s);
      const float fac   = __expf(m_run - m_new);   // 0 when m_run == -inf
      const float w     = __expf(s - m_new);
      l_run = l_run * fac + w;
      acc.x = acc.x * fac + w * row[r].x;
      acc.y = acc.y * fac + w * row[r].y;
      acc.z = acc.z * fac + w * row[r].z;
      acc.w = acc.w * fac + w * row[r].w;
      m_run = m_new;
    }
    __syncthreads();   // protect part[][] before next group's writes
  }

  *(v4f*)(ws + WS_ACC + (size_t)blockIdx.x * H_ + 4 * t) = acc;
  if (t == 0) {
    ws[WS_M + blockIdx.x] = m_run;
    ws[WS_L + blockIdx.x] = l_run;
  }
}

// ---------------------------------------------------------------------------
// Kernel 2: merge the C_ chunk partials per batch (flash-merge), normalize,
// and emit combined = [hidden | attn_applied]  ([B, 2H]) into workspace.
// ---------------------------------------------------------------------------
__global__ __launch_bounds__(TPB) void attn_merge(
    const float* __restrict__ hidden, float* __restrict__ ws)
{
  const int t = threadIdx.x;
  const int b = blockIdx.x;

  float M = -INFINITY;
#pragma unroll
  for (int c = 0; c < C_; ++c) M = fmaxf(M, ws[WS_M + b * C_ + c]);

  float f[C_];
  float L = 0.f;
#pragma unroll
  for (int c = 0; c < C_; ++c) {
    f[c] = __expf(ws[WS_M + b * C_ + c] - M);
    L   += ws[WS_L + b * C_ + c] * f[c];
  }

  v4f a = {0.f, 0.f, 0.f, 0.f};
#pragma unroll
  for (int c = 0; c < C_; ++c) {
    const v4f p = *(const v4f*)(ws + WS_ACC + (size_t)(b * C_ + c) * H_ + 4 * t);
    a.x += f[c] * p.x;  a.y += f[c] * p.y;
    a.z += f[c] * p.z;  a.w += f[c] * p.w;
  }
  const float inv = 1.f / L;
  a.x *= inv; a.y *= inv; a.z *= inv; a.w *= inv;

  const v4f h4 = *(const v4f*)(hidden + (size_t)b * H_ + 4 * t);
  float* comb = ws + WS_COMB + (size_t)b * (2 * H_);
  *(v4f*)(comb + 4 * t)      = h4;   // combined[b, 0:H]   = hidden
  *(v4f*)(comb + H_ + 4 * t) = a;    // combined[b, H:2H]  = attn_applied
}

// ---------------------------------------------------------------------------
// Kernel 3: out[64,1024] = combined[64,2048] @ W^T[2048,1024] + bias
// via V_WMMA_F32_16X16X4_F32. One wave per 16x16 output tile.
//   A frag (16x4):  vgpr = k&1, lane = m + 16*(k>>1)  -> contiguous float2
//   B frag (4x16):  vgpr = k&1, lane = n + 16*(k>>1); B[k,n] = W[n,k]
//   C/D: vgpr i -> M = i + 8*(lane>>4), N = lane&15
// ---------------------------------------------------------------------------
__global__ __launch_bounds__(TPB) void gemm_wmma(
    const float* __restrict__ comb, const float* __restrict__ W,
    const float* __restrict__ bias, float* __restrict__ out)
{
  const int lane = threadIdx.x & 31;
  const int wid  = threadIdx.x >> 5;
  const int gw   = blockIdx.x * (TPB / 32) + wid;  // 0..255 tiles
  const int m0   = (gw >> 6) * 16;                 // 4 M-tiles
  const int n0   = (gw & 63) * 16;                 // 64 N-tiles
  const int half = lane >> 4;
  const int r    = lane & 15;

  const float* aptr = comb + (size_t)(m0 + r) * (2 * H_) + 2 * half;
  const float* bptr = W    + (size_t)(n0 + r) * (2 * H_) + 2 * half;

  v8f acc = {};
#pragma unroll 8
  for (int kk = 0; kk < 2 * H_; kk += 4) {
    const v2f av = *(const v2f*)(aptr + kk);
    const v2f bv = *(const v2f*)(bptr + kk);
    acc = __builtin_amdgcn_wmma_f32_16x16x4_f32(
        /*neg_a=*/false, av, /*neg_b=*/false, bv,
        /*c_mod=*/(short)0, acc, /*reuse_a=*/false, /*reuse_b=*/false);
  }

  const float bb = bias[n0 + r];
#pragma unroll
  for (int i = 0; i < 8; ++i) {
    const int row = m0 + half * 8 + i;
    out[(size_t)row * H_ + n0 + r] = acc[i] + bb;
  }
}

// ---------------------------------------------------------------------------
extern "C" void kernel_launch(void* const* d_in, const int* in_sizes, int n_in,
                              void* d_out, int out_size, void* d_ws, size_t ws_size,
                              hipStream_t stream) {
  const float* lstm   = (const float*)d_in[0];
  const float* hidden = (const float*)d_in[1];
  const float* W      = (const float*)d_in[2];
  const float* bias   = (const float*)d_in[3];
  float* out = (float*)d_out;
  float* ws  = (float*)d_ws;

  attn_pass1<<<B_ * C_, TPB, 0, stream>>>(lstm, hidden, ws);
  attn_merge<<<B_, TPB, 0, stream>>>(hidden, ws);
  // 256 output tiles / 8 waves per block = 32 blocks
  gemm_wmma<<<32, TPB, 0, stream>>>(ws + WS_COMB, W, bias, out);
}